// ShapeConsistencyLoss_25074019074110
// MI455X (gfx1250) — compile-verified
//
#include <hip/hip_runtime.h>
#include <math.h>

typedef float v2f_t __attribute__((ext_vector_type(2)));
typedef float v8f_t __attribute__((ext_vector_type(8)));

namespace {

constexpr int   kB        = 4;
constexpr int   kN        = 4096;
constexpr int   kTiles    = kN / 16;   // 256 sixteen-point tiles per cloud
constexpr int   kPairs    = kTiles / 2;
constexpr float kSelfMask = 1e6f;

__global__ void init_out(float* out) { out[0] = 0.0f; }

__device__ inline float block_reduce_sum(float v, float* buf) {
  const int tid = threadIdx.x;
  buf[tid] = v;
  __syncthreads();
  for (int s = 128; s > 0; s >>= 1) {
    if (tid < s) buf[tid] += buf[tid + s];
    __syncthreads();
  }
  const float r = buf[0];
  __syncthreads();
  return r;
}

// ---- Part 1: unbiased variance of distances to centroid, per cloud ----
__global__ __launch_bounds__(256) void var_kernel(const float* __restrict__ pc,
                                                  float* __restrict__ out) {
  __shared__ float buf[256];
  const int b   = blockIdx.x;
  const int tid = threadIdx.x;
  const float* base = pc + (size_t)b * kN * 3;

  float sx = 0.f, sy = 0.f, sz = 0.f;
  for (int p = tid; p < kN; p += 256) {
    sx += base[p * 3 + 0];
    sy += base[p * 3 + 1];
    sz += base[p * 3 + 2];
  }
  const float cx = block_reduce_sum(sx, buf) * (1.0f / kN);
  const float cy = block_reduce_sum(sy, buf) * (1.0f / kN);
  const float cz = block_reduce_sum(sz, buf) * (1.0f / kN);

  float sd = 0.f, sd2 = 0.f;
  for (int p = tid; p < kN; p += 256) {
    const float dx = base[p * 3 + 0] - cx;
    const float dy = base[p * 3 + 1] - cy;
    const float dz = base[p * 3 + 2] - cz;
    const float d2 = dx * dx + dy * dy + dz * dz;
    sd  += sqrtf(d2);
    sd2 += d2;
  }
  const float tsd  = block_reduce_sum(sd, buf);
  const float tsd2 = block_reduce_sum(sd2, buf);
  if (tid == 0) {
    const float var = (tsd2 - tsd * tsd * (1.0f / kN)) * (1.0f / (kN - 1));
    atomicAdd(out, var * (1.0f / kB));
  }
}

// Raw CDNA5 min ops (inputs are finite -> skip fminf's NaN canonicalization).
__device__ __forceinline__ float vmin_raw(float a, float b) {
  float r;
  asm("v_min_num_f32 %0, %1, %2" : "=v"(r) : "v"(a), "v"(b));
  return r;
}
__device__ __forceinline__ float vmin3_raw(float a, float b, float c) {
  float r;
  asm("v_min3_num_f32 %0, %1, %2, %3" : "=v"(r) : "v"(a), "v"(b), "v"(c));
  return r;
}

// XOR-butterfly min within each 16-lane half via ds_swizzle (group-of-32:
// offset = xor_mask<<10 | and_mask 0x1f). Matches the D-matrix half-wave layout.
__device__ __forceinline__ float half_min16(float v) {
  float o;
  o = __builtin_bit_cast(float, __builtin_amdgcn_ds_swizzle(__builtin_bit_cast(int, v), 0x041F));
  v = vmin_raw(v, o);
  o = __builtin_bit_cast(float, __builtin_amdgcn_ds_swizzle(__builtin_bit_cast(int, v), 0x081F));
  v = vmin_raw(v, o);
  o = __builtin_bit_cast(float, __builtin_amdgcn_ds_swizzle(__builtin_bit_cast(int, v), 0x101F));
  v = vmin_raw(v, o);
  o = __builtin_bit_cast(float, __builtin_amdgcn_ds_swizzle(__builtin_bit_cast(int, v), 0x201F));
  v = vmin_raw(v, o);
  return v;
}

// One J-tile PAIR (tiles 2p and 2p+1): D = A x B + 0 with
// A row i = [-2xi,-2yi,-2zi,1], B col j = [xj,yj,zj,|pj|^2]
//   =>  D[i][j] = |pj|^2 - 2 pi.pj
// (|pi|^2 is row-constant -> added after the column-min; clamp deferred too.)
// bsrc is the half-selected packed B operand: lanes 0-15 {x,y}, lanes 16-31 {z,sq}.
// MASKED: this is the wave's own pair (tile 2p == it0, tile 2p+1 == it1);
// mask the self-distance of acc0 in tile-a and of acc1 in tile-b.
template <bool MASKED>
__device__ __forceinline__ void pair_step(const float2* __restrict__ bsrc,
                                          int p, int col, int rowoff,
                                          v2f_t a0, v2f_t a1,
                                          float m0[8], float m1[8]) {
  const float2 ta = bsrc[(2 * p) * 16 + col];       // merge: ds_load_2addr_b64
  const float2 tb = bsrc[(2 * p + 1) * 16 + col];
  v2f_t ba, bb;
  ba.x = ta.x; ba.y = ta.y;
  bb.x = tb.x; bb.y = tb.y;

  v8f_t cz = {};                                    // inline 0 accumulator
  v8f_t d0a = __builtin_amdgcn_wmma_f32_16x16x4_f32(
      false, a0, false, ba, (short)0, cz, false, false);
  v8f_t d1a = __builtin_amdgcn_wmma_f32_16x16x4_f32(
      false, a1, false, ba, (short)0, cz, false, false);
  v8f_t d0b = __builtin_amdgcn_wmma_f32_16x16x4_f32(
      false, a0, false, bb, (short)0, cz, false, false);
  v8f_t d1b = __builtin_amdgcn_wmma_f32_16x16x4_f32(
      false, a1, false, bb, (short)0, cz, false, false);

#pragma unroll
  for (int v = 0; v < 8; ++v) {
    float e0a = d0a[v], e0b = d0b[v];
    float e1a = d1a[v], e1b = d1b[v];
    if (MASKED && (v + rowoff) == col) {
      e0a += kSelfMask;   // acc0's diagonal lives in tile-a (jt = it0)
      e1b += kSelfMask;   // acc1's diagonal lives in tile-b (jt = it1)
    }
    m0[v] = vmin3_raw(m0[v], e0a, e0b);
    m1[v] = vmin3_raw(m1[v], e1a, e1b);
  }
}

// ---- Part 2: nearest-neighbor repulsion via v_wmma_f32_16x16x4_f32 ----
__global__ __launch_bounds__(256) void nn_kernel(const float* __restrict__ pc,
                                                 float* __restrict__ out) {
  __shared__ float2 sblo[kN];   // {x, y}     : B rows 0,1 (lanes 0-15)
  __shared__ float2 sbhi[kN];   // {z, |p|^2} : B rows 2,3 (lanes 16-31)

  const int tid    = threadIdx.x;
  const int b      = blockIdx.x >> 4;                        // 16 blocks/cloud
  const int pw     = __builtin_amdgcn_readfirstlane(
                       ((blockIdx.x & 15) << 3) + (tid >> 5)); // wave pair id
  const int it0    = 2 * pw;           // adjacent I-tiles: both diagonals in pair pw
  const int it1    = 2 * pw + 1;
  const int lane   = tid & 31;
  const int col    = lane & 15;
  const bool hi    = lane >= 16;
  const int rowoff = hi ? 8 : 0;

  const float* base = pc + (size_t)b * kN * 3;
  for (int p = tid; p < kN; p += 256) {
    const float x = base[p * 3 + 0];
    const float y = base[p * 3 + 1];
    const float z = base[p * 3 + 2];
    sblo[p] = make_float2(x, y);
    sbhi[p] = make_float2(z, x * x + y * y + z * z);
  }
  __syncthreads();

  const float2* bsrc = hi ? sbhi : sblo;    // half-select once, outside the loop

  // A operands: row i = [-2xi, -2yi, -2zi, 1]
  // lanes 0-15 hold K=0,1 (=-2x,-2y); lanes 16-31 hold K=2,3 (=-2z, 1)
  const float2 t0 = bsrc[it0 * 16 + col];
  const float2 t1 = bsrc[it1 * 16 + col];
  v2f_t a0, a1;
  a0.x = -2.0f * t0.x;
  a0.y = hi ? 1.0f : (-2.0f * t0.y);
  a1.x = -2.0f * t1.x;
  a1.y = hi ? 1.0f : (-2.0f * t1.y);

  float m0[8], m1[8];
#pragma unroll
  for (int v = 0; v < 8; ++v) { m0[v] = 3.0e38f; m1[v] = 3.0e38f; }

  // Guard-free pair sweep; the wave's own (diagonal) pair is peeled.
#pragma unroll 2
  for (int p = 0; p < pw; ++p)
    pair_step<false>(bsrc, p, col, rowoff, a0, a1, m0, m1);
  pair_step<true>(bsrc, pw, col, rowoff, a0, a1, m0, m1);
#pragma unroll 2
  for (int p = pw + 1; p < kPairs; ++p)
    pair_step<false>(bsrc, p, col, rowoff, a0, a1, m0, m1);

  // Column-min within each 16-lane half; all WMMAs are done at this point.
#pragma unroll
  for (int v = 0; v < 8; ++v) {
    m0[v] = half_min16(m0[v]);
    m1[v] = half_min16(m1[v]);
  }

  if (col == 0) {  // lane 0 -> rows 0-7, lane 16 -> rows 8-15 of each tile
    float partial = 0.0f;
#pragma unroll
    for (int v = 0; v < 8; ++v) {
      const int row = rowoff + v;
      const float sq0 = sbhi[it0 * 16 + row].y;   // stored |p_i|^2
      const float sq1 = sbhi[it1 * 16 + row].y;
      partial += expf(-5.0f * fmaxf(m0[v] + sq0, 0.0f));
      partial += expf(-5.0f * fmaxf(m1[v] + sq1, 0.0f));
    }
    atomicAdd(out, partial * (1.0f / (kB * kN)));
  }
}

}  // namespace

extern "C" void kernel_launch(void* const* d_in, const int* in_sizes, int n_in,
                              void* d_out, int out_size, void* d_ws, size_t ws_size,
                              hipStream_t stream) {
  (void)in_sizes; (void)n_in; (void)out_size; (void)d_ws; (void)ws_size;
  const float* pc  = (const float*)d_in[0];
  float*       out = (float*)d_out;

  hipLaunchKernelGGL(init_out, dim3(1), dim3(1), 0, stream, out);
  hipLaunchKernelGGL(var_kernel, dim3(kB), dim3(256), 0, stream, pc, out);
  hipLaunchKernelGGL(nn_kernel, dim3(kB * 16), dim3(256), 0, stream, pc, out);
}